// PointNetSetAbstraction_60017872994423
// MI455X (gfx1250) — compile-verified
//
#include <hip/hip_runtime.h>
#include <hip/hip_bf16.h>
#include <stdint.h>

// ---------------------------------------------------------------------------
// PointNet++ Set Abstraction for MI455X (gfx1250, wave32).
// FPS (serial, latency-bound) -> ball query (wave32 ballot compaction) ->
// grouping -> 3x [f32 WMMA GEMM -> batchnorm stats -> norm+relu] -> max over K.
// MLP is bandwidth-bound (13.4 GFLOP vs ~1.6GB traffic @ 23.3TB/s), so we use
// the exact-precision V_WMMA_F32_16X16X4_F32 path instead of f16 WMMA.
// GEMM tiles are staged into LDS by the Tensor Data Mover when the toolchain
// exposes it (descriptor built per CDNA5 ISA), else by a
// vectorized b128 copy.
// ---------------------------------------------------------------------------

typedef __attribute__((ext_vector_type(2))) float v2f;
typedef __attribute__((ext_vector_type(4))) float v4f;
typedef __attribute__((ext_vector_type(8))) float v8f;
typedef __attribute__((ext_vector_type(4))) unsigned int u32x4;
typedef __attribute__((ext_vector_type(4))) int i32x4;
typedef __attribute__((ext_vector_type(8))) int i32x8;

#define BB 16
#define NN 4096
#define SS 1024
#define KK 32
#define R2 0.16f
#define MROWS ((size_t)BB * SS * KK) /* 524288 */

#if __has_builtin(__builtin_amdgcn_tensor_load_to_lds) && \
    __has_builtin(__builtin_amdgcn_s_wait_tensorcnt)
#define GEMM_TDM 1
#endif

// ---------------------------------------------------------------------------
// Farthest point sampling: one block per batch, 1024 threads (32 waves).
// dist[] lives in LDS; per step: per-thread min-update + two-level argmax
// (wave shuffle reduce, then wave0 reduces 32 partials). Tie-break = lowest
// index to match jnp.argmax. Also emits new_xyz [B,S,3] and the transposed
// xyz_query output [B,3,S].
// ---------------------------------------------------------------------------
__global__ __launch_bounds__(1024) void fps_kernel(
    const float* __restrict__ xyz_pc, float* __restrict__ new_xyz,
    float* __restrict__ out_xyzq) {
  const int b = blockIdx.x;
  const int tid = threadIdx.x;
  const int lane = tid & 31, wid = tid >> 5;
  __shared__ float dist[NN];
  __shared__ float wv[32];
  __shared__ int wi[32];
  __shared__ int fsh;
  const float* px = xyz_pc + (size_t)b * 3 * NN;
  const float* py = px + NN;
  const float* pz = py + NN;
  for (int i = tid; i < NN; i += 1024) dist[i] = 1e10f;
  __syncthreads();
  int f = 0;
  for (int s = 0; s < SS; ++s) {
    const float cx = px[f], cy = py[f], cz = pz[f];
    if (tid == 0) {
      const size_t nb = ((size_t)b * SS + s) * 3;
      new_xyz[nb + 0] = cx;
      new_xyz[nb + 1] = cy;
      new_xyz[nb + 2] = cz;
      const size_t ob = (size_t)b * 3 * SS + s;
      out_xyzq[ob] = cx;
      out_xyzq[ob + SS] = cy;
      out_xyzq[ob + 2 * SS] = cz;
    }
    float best = -1.0f;
    int bi = 0;
    for (int i = tid; i < NN; i += 1024) {
      const float dx = px[i] - cx, dy = py[i] - cy, dz = pz[i] - cz;
      const float d = dx * dx + dy * dy + dz * dz;
      const float nd = fminf(dist[i], d);
      dist[i] = nd;
      if (nd > best) { best = nd; bi = i; }
    }
#pragma unroll
    for (int off = 16; off > 0; off >>= 1) {
      const float ov = __shfl_down(best, off);
      const int oi = __shfl_down(bi, off);
      if (ov > best || (ov == best && oi < bi)) { best = ov; bi = oi; }
    }
    if (lane == 0) { wv[wid] = best; wi[wid] = bi; }
    __syncthreads();
    if (wid == 0) {
      best = wv[lane];
      bi = wi[lane];
#pragma unroll
      for (int off = 16; off > 0; off >>= 1) {
        const float ov = __shfl_down(best, off);
        const int oi = __shfl_down(bi, off);
        if (ov > best || (ov == best && oi < bi)) { best = ov; bi = oi; }
      }
      if (lane == 0) fsh = bi;
    }
    __syncthreads();
    f = fsh;
  }
}

// ---------------------------------------------------------------------------
// Ball query: one wave32 per query. Ordered first-K compaction via ballot +
// prefix popcount (equivalent to the reference's where/sort/take-K). Pads
// with the first in-radius index (always exists: the query point itself).
// ---------------------------------------------------------------------------
__global__ __launch_bounds__(256) void ballq_kernel(
    const float* __restrict__ xyz_pc, const float* __restrict__ new_xyz,
    int* __restrict__ gidx) {
  const int q = blockIdx.x * (blockDim.x >> 5) + (threadIdx.x >> 5);
  if (q >= BB * SS) return;
  const int lane = threadIdx.x & 31;
  const int b = q >> 10;  // / SS
  const float* px = xyz_pc + (size_t)b * 3 * NN;
  const float* py = px + NN;
  const float* pz = py + NN;
  const float qx = new_xyz[(size_t)q * 3 + 0];
  const float qy = new_xyz[(size_t)q * 3 + 1];
  const float qz = new_xyz[(size_t)q * 3 + 2];
  int* gq = gidx + (size_t)q * KK;
  int cnt = 0, first = 0;
  bool haveFirst = false;
  for (int c0 = 0; c0 < NN; c0 += 32) {
    const int i = c0 + lane;
    const float dx = px[i] - qx, dy = py[i] - qy, dz = pz[i] - qz;
    const bool hit = (dx * dx + dy * dy + dz * dz) <= R2;
    const unsigned int bal = (unsigned int)__ballot(hit);
    if (!haveFirst && bal) { first = c0 + (__ffs(bal) - 1); haveFirst = true; }
    if (hit) {
      const int pos = cnt + __popc(bal & ((1u << lane) - 1u));
      if (pos < KK) gq[pos] = i;
    }
    cnt += __popc(bal);
    if (cnt >= KK) break;
  }
  for (int j = cnt + lane; j < KK; j += 32) gq[j] = first;
}

// ---------------------------------------------------------------------------
// Build X0 [M, 8] (6 real channels: relative xyz + feats, padded to 8).
// ---------------------------------------------------------------------------
__global__ __launch_bounds__(256) void group_kernel(
    const float* __restrict__ xyz_pc, const float* __restrict__ feats_pc,
    const float* __restrict__ new_xyz, const int* __restrict__ gidx,
    float* __restrict__ X0) {
  const size_t m = (size_t)blockIdx.x * blockDim.x + threadIdx.x;
  if (m >= MROWS) return;
  const int g = gidx[m];
  const int sk = SS * KK;
  const int b = (int)(m / sk);
  const int r = (int)(m % sk);
  const int s = r / KK;
  const float* px = xyz_pc + (size_t)b * 3 * NN;
  const float* fx = feats_pc + (size_t)b * 3 * NN;
  const size_t nb = ((size_t)b * SS + s) * 3;
  float* o = X0 + m * 8;
  o[0] = px[g] - new_xyz[nb + 0];
  o[1] = px[NN + g] - new_xyz[nb + 1];
  o[2] = px[2 * NN + g] - new_xyz[nb + 2];
  o[3] = fx[g];
  o[4] = fx[NN + g];
  o[5] = fx[2 * NN + g];
  o[6] = 0.0f;
  o[7] = 0.0f;
}

#ifdef GEMM_TDM
// Build a D# per CDNA5 ISA and issue TENSOR_LOAD_TO_LDS.
// 2-D tile (tile_d0 x tile_d1) of 4-byte elements; rows of tile_d0 DWORDs are
// placed at (tile_d0 + pad) DWORD pitch in LDS via pad_enable; x >= tensor_d0
// reads return zero (free K-padding). lds_off is a byte offset from LDS base 0
// (single __shared__ array in the kernel => deterministic offsets).
__device__ __forceinline__ void tdm_load_tile(
    unsigned lds_off, const void* gaddr, unsigned tensor_d0, unsigned tensor_d1,
    unsigned tile_d0, unsigned tile_d1, unsigned stride0, unsigned pad_interval,
    unsigned pad_amount) {
  const unsigned long long ga = (unsigned long long)gaddr;
  u32x4 g0;
  g0[0] = 1u;  // count=1 (valid user descriptor), gather off
  g0[1] = lds_off;
  g0[2] = (unsigned)ga;
  g0[3] = (unsigned)((ga >> 32) & 0x1FFFFFFu) | (2u << 30);  // type=2 (image)
  i32x8 g1;
  g1[0] = (int)((2u << 16) |            // data_size = 4 bytes
                (1u << 20) |            // pad_enable
                (pad_interval << 22) | (pad_amount << 25));
  g1[1] = (int)((tensor_d0 & 0xFFFFu) << 16);           // tensor_dim0 lo16
  g1[2] = (int)((tensor_d0 >> 16) | ((tensor_d1 & 0xFFFFu) << 16));
  g1[3] = (int)((tensor_d1 >> 16) | (tile_d0 << 16));   // tile_dim0
  g1[4] = (int)tile_d1;                                 // tile_dim1 (dim2=0)
  g1[5] = (int)stride0;                                 // dim0_stride lo32
  g1[6] = 0;
  g1[7] = 0;
  const i32x4 z4 = {0, 0, 0, 0};
#if __clang_major__ >= 23
  const i32x8 z8 = {0, 0, 0, 0, 0, 0, 0, 0};
  __builtin_amdgcn_tensor_load_to_lds(g0, g1, z4, z4, z8, 0);
#else
  __builtin_amdgcn_tensor_load_to_lds(g0, g1, z4, z4, 0);
#endif
}
#endif

// ---------------------------------------------------------------------------
// f32 WMMA GEMM: Y[M,O] = X[M,LDX] * W[O,CW]^T + bias. One wave32 per 16x16
// output tile, fully-unrolled K-loop of V_WMMA_F32_16X16X4_F32.
//  * A-tile: per-wave contiguous 16*LDX region -> LDS (TDM or b128 copy).
//  * B-tile: all 8 waves in a block share ntile -> staged once per block,
//    zero-padded for k >= CW (TDM does this via tensor_dim0 = CW).
//  * PITCH = LDX+4: rows 16B-aligned (b128 stores / TDM pad of 4 DWORDs) and
//    frag reads are 8B ds_load_b64 hitting distinct banks.
// Fragment layouts per CDNA5 ISA:
//  A 16x4 f32: lane L holds row L&15; half-wave selects K pair
//    (VGPR0=K{0,2}, VGPR1=K{1,3}).  B mirrored.  C/D: 8 VGPRs,
//    row = v + 8*(L>=16), col = L&15.
// Grid exactly covers mtiles*(O/16) tiles (mtiles % 8 == 0), so every wave
// is active and the __syncthreads() is safe.
// ---------------------------------------------------------------------------
template <int LDX, int CW>
__global__ __launch_bounds__(256) void gemm_wmma_f32(
    const float* __restrict__ X, const float* __restrict__ W,
    const float* __restrict__ bias, float* __restrict__ Y, int O, int mtiles) {
  constexpr int KSTEPS = LDX / 4;
  constexpr int PITCH = LDX + 4;
  constexpr int LSH = (LDX == 64) ? 6 : 3;  // log2(LDX)
  // single shared object => LDS base offset 0 (TDM descriptors rely on this)
  __shared__ __align__(16) float smem[(8 * 16 + 16) * PITCH];
  float* const Asw = smem + (threadIdx.x >> 5) * 16 * PITCH;
  float* const Bs = smem + 8 * 16 * PITCH;
  const int lane = threadIdx.x & 31;
  const int w = threadIdx.x >> 5;
  const int mblk = mtiles >> 3;  // blocks per ntile stripe
  const int ntile = blockIdx.x / mblk;
  const int mtile = (blockIdx.x % mblk) * 8 + w;
  const float* at = X + (size_t)mtile * 16 * LDX;

#ifdef GEMM_TDM
  constexpr unsigned PI = (LDX == 64) ? 5u : 2u;  // pad every LDX DWORDs
  constexpr unsigned PA = 3u;                     // pad 4 DWORDs (= PITCH-LDX)
  // per-wave A tile: 1-D run of 16*LDX DWORDs, LDS pitch padding via TDM
  tdm_load_tile((unsigned)(w * 16 * PITCH * 4), at, 16 * LDX, 1, 16 * LDX, 1,
                16 * LDX, PI, PA);
  if (w == 0) {
    // shared B tile: 16 rows x LDX, zero-fill for k >= CW via tensor_dim0=CW
    tdm_load_tile((unsigned)(8 * 16 * PITCH * 4), W + (size_t)ntile * 16 * CW,
                  CW, 16, LDX, 16, CW, PI, PA);
  }
  __builtin_amdgcn_s_wait_tensorcnt(0);
#else
  // stage A: per-wave b128 copy of the contiguous 16*LDX tile
#pragma unroll
  for (int i = 0; i < (16 * LDX) / 128; ++i) {
    const int e4 = (i * 32 + lane) * 4;
    const int r = e4 >> LSH;
    const int c = e4 & (LDX - 1);
    *(v4f*)&Asw[r * PITCH + c] = *(const v4f*)&at[e4];
  }
  // stage B: block-cooperative
  if constexpr (CW == LDX) {
    const int e4 = threadIdx.x * 4;
    if (e4 < 16 * LDX) {
      const int col = e4 >> LSH;
      const int k = e4 & (LDX - 1);
      *(v4f*)&Bs[col * PITCH + k] =
          *(const v4f*)&W[((size_t)ntile * 16 + col) * CW + k];
    }
  } else {
    for (int idx = threadIdx.x; idx < 16 * LDX; idx += 256) {
      const int col = idx >> LSH;
      const int k = idx & (LDX - 1);
      Bs[col * PITCH + k] =
          (k < CW) ? W[((size_t)ntile * 16 + col) * CW + k] : 0.0f;
    }
  }
#endif
  __syncthreads();

  const int arow = lane & 15;
  const int khalf = (lane >> 4) << 1;
  const int col = lane & 15;
  v8f acc = {};
#pragma unroll
  for (int ks = 0; ks < KSTEPS; ++ks) {
    const int kb = ks * 4 + khalf;  // even -> 8B-aligned LDS reads
    const v2f a = *(const v2f*)&Asw[arow * PITCH + kb];
    const v2f bm = *(const v2f*)&Bs[col * PITCH + kb];
    acc = __builtin_amdgcn_wmma_f32_16x16x4_f32(false, a, false, bm, (short)0,
                                                acc, false, false);
  }
  const int wrow = ntile * 16 + col;
  const float bv = bias[wrow];
  const int rbase = mtile * 16 + ((lane >> 4) << 3);
#pragma unroll
  for (int v = 0; v < 8; ++v) {
    Y[(size_t)(rbase + v) * O + wrow] = acc[v] + bv;
  }
}

// ---------------------------------------------------------------------------
// Per-channel sum / sum-of-squares over all M rows (coalesced: 256 threads
// cover 256/O consecutive rows per step), LDS tree, then O atomics per block.
// ---------------------------------------------------------------------------
__global__ __launch_bounds__(256) void stats_kernel(
    const float* __restrict__ Y, int O, int oShift, size_t M,
    float* __restrict__ gsum, float* __restrict__ gsq) {
  __shared__ float s1[256], s2[256];
  const int t = threadIdx.x;
  const int o = t & (O - 1);
  const int rsub = t >> oShift;
  const int rpi = 256 >> oShift;
  float a1 = 0.0f, a2 = 0.0f;
  for (size_t row = (size_t)blockIdx.x * rpi + rsub; row < M;
       row += (size_t)gridDim.x * rpi) {
    const float v = Y[row * O + o];
    a1 += v;
    a2 += v * v;
  }
  s1[t] = a1;
  s2[t] = a2;
  __syncthreads();
  if (t < O) {
    for (int j = t + O; j < 256; j += O) { a1 += s1[j]; a2 += s2[j]; }
    atomicAdd(&gsum[o], a1);
    atomicAdd(&gsq[o], a2);
  }
}

__global__ __launch_bounds__(128) void bnfin_kernel(
    const float* __restrict__ gsum, const float* __restrict__ gsq,
    const float* __restrict__ g, const float* __restrict__ be,
    float* __restrict__ scale, float* __restrict__ shift, int O, float invM) {
  const int o = threadIdx.x;
  if (o >= O) return;
  const float mean = gsum[o] * invM;
  const float var = gsq[o] * invM - mean * mean;
  const float s = rsqrtf(var + 1e-5f) * g[o];
  scale[o] = s;
  shift[o] = be[o] - mean * s;
}

__global__ __launch_bounds__(256) void normrelu_kernel(
    float* __restrict__ Y, const float* __restrict__ scale,
    const float* __restrict__ shift, int O, size_t total) {
  const size_t i = (size_t)blockIdx.x * blockDim.x + threadIdx.x;
  if (i >= total) return;
  const int o = (int)(i & (size_t)(O - 1));
  const float v = Y[i] * scale[o] + shift[o];
  Y[i] = v > 0.0f ? v : 0.0f;
}

// ---------------------------------------------------------------------------
// Max over K neighbors + transpose to [B, 128, S].
// ---------------------------------------------------------------------------
__global__ __launch_bounds__(128) void maxk_kernel(
    const float* __restrict__ Y3, float* __restrict__ outp) {
  const int bs = blockIdx.x;
  const int o = threadIdx.x;
  const int b = bs >> 10, s = bs & 1023;
  const float* base = Y3 + (size_t)bs * KK * 128 + o;
  float m = base[0];
#pragma unroll 4
  for (int k = 1; k < KK; ++k) m = fmaxf(m, base[(size_t)k * 128]);
  outp[((size_t)b * 128 + o) * SS + s] = m;
}

// ---------------------------------------------------------------------------
extern "C" void kernel_launch(void* const* d_in, const int* in_sizes, int n_in,
                              void* d_out, int out_size, void* d_ws,
                              size_t ws_size, hipStream_t stream) {
  const float* xyz_pc = (const float*)d_in[0];
  const float* points = (const float*)d_in[1];
  const float* W1 = (const float*)d_in[2];
  const float* b1 = (const float*)d_in[3];
  const float* g1 = (const float*)d_in[4];
  const float* be1 = (const float*)d_in[5];
  const float* W2 = (const float*)d_in[6];
  const float* b2 = (const float*)d_in[7];
  const float* g2 = (const float*)d_in[8];
  const float* be2 = (const float*)d_in[9];
  const float* W3 = (const float*)d_in[10];
  const float* b3 = (const float*)d_in[11];
  const float* g3 = (const float*)d_in[12];
  const float* be3 = (const float*)d_in[13];
  (void)in_sizes; (void)n_in; (void)out_size; (void)ws_size;

  // workspace carve (ping/pong reuse: X0->ping, Y1->pong, Y2->ping, Y3->pong)
  char* ws = (char*)d_ws;
  size_t off = 0;
  auto carve = [&](size_t bytes) -> char* {
    char* p = ws + off;
    off += (bytes + 255) & ~(size_t)255;
    return p;
  };
  float* new_xyz = (float*)carve((size_t)BB * SS * 3 * 4);
  int* gidx = (int*)carve((size_t)BB * SS * KK * 4);
  float* gsum = (float*)carve(128 * 4);
  float* gsq = (float*)carve(128 * 4);
  float* scale = (float*)carve(128 * 4);
  float* shift = (float*)carve(128 * 4);
  float* ping = (float*)carve(MROWS * 64 * 4);   // X0 (M*8), later Y2 (M*64)
  float* pong = (float*)carve(MROWS * 128 * 4);  // Y1 (M*64), later Y3 (M*128)

  float* out_xyzq = (float*)d_out;                       // [B,3,S]
  float* out_pts = (float*)d_out + (size_t)BB * 3 * SS;  // [B,128,S]

  fps_kernel<<<BB, 1024, 0, stream>>>(xyz_pc, new_xyz, out_xyzq);
  ballq_kernel<<<(BB * SS) / 8, 256, 0, stream>>>(xyz_pc, new_xyz, gidx);
  group_kernel<<<(int)(MROWS / 256), 256, 0, stream>>>(xyz_pc, points, new_xyz,
                                                       gidx, ping);

  const int mtiles = (int)(MROWS / 16);  // 32768, divisible by 8
  const int mblk = mtiles / 8;

  // ---- layer 1: [M,6(->8)] x W1[64,6]^T -> Y1[M,64] ----
  hipMemsetAsync(gsum, 0, 128 * 4, stream);
  hipMemsetAsync(gsq, 0, 128 * 4, stream);
  gemm_wmma_f32<8, 6><<<mblk * 4, 256, 0, stream>>>(ping, W1, b1, pong, 64,
                                                    mtiles);
  stats_kernel<<<2048, 256, 0, stream>>>(pong, 64, 6, MROWS, gsum, gsq);
  bnfin_kernel<<<1, 128, 0, stream>>>(gsum, gsq, g1, be1, scale, shift, 64,
                                      1.0f / (float)MROWS);
  normrelu_kernel<<<(int)(MROWS * 64 / 256), 256, 0, stream>>>(
      pong, scale, shift, 64, MROWS * 64);

  // ---- layer 2: Y1[M,64] x W2[64,64]^T -> Y2[M,64] ----
  hipMemsetAsync(gsum, 0, 128 * 4, stream);
  hipMemsetAsync(gsq, 0, 128 * 4, stream);
  gemm_wmma_f32<64, 64><<<mblk * 4, 256, 0, stream>>>(pong, W2, b2, ping, 64,
                                                      mtiles);
  stats_kernel<<<2048, 256, 0, stream>>>(ping, 64, 6, MROWS, gsum, gsq);
  bnfin_kernel<<<1, 128, 0, stream>>>(gsum, gsq, g2, be2, scale, shift, 64,
                                      1.0f / (float)MROWS);
  normrelu_kernel<<<(int)(MROWS * 64 / 256), 256, 0, stream>>>(
      ping, scale, shift, 64, MROWS * 64);

  // ---- layer 3: Y2[M,64] x W3[128,64]^T -> Y3[M,128] ----
  hipMemsetAsync(gsum, 0, 128 * 4, stream);
  hipMemsetAsync(gsq, 0, 128 * 4, stream);
  gemm_wmma_f32<64, 64><<<mblk * 8, 256, 0, stream>>>(ping, W3, b3, pong, 128,
                                                      mtiles);
  stats_kernel<<<2048, 256, 0, stream>>>(pong, 128, 7, MROWS, gsum, gsq);
  bnfin_kernel<<<1, 128, 0, stream>>>(gsum, gsq, g3, be3, scale, shift, 128,
                                      1.0f / (float)MROWS);
  normrelu_kernel<<<(int)(MROWS * 128 / 256), 256, 0, stream>>>(
      pong, scale, shift, 128, MROWS * 128);

  maxk_kernel<<<BB * SS, 128, 0, stream>>>(pong, out_pts);
}